// Net_11141145166043
// MI455X (gfx1250) — compile-verified
//
#include <hip/hip_runtime.h>
#include <math.h>

// ---------------- problem constants ----------------
#define NN      10000
#define NE      160000
#define IN_CH   1433
#define HID     16
#define OUTC    7
#define NCAT    48          // [W1_spline0 | W1_spline1 | W1_root] -> 48 cols
#define KPAD    1440        // 1433 padded up to multiple of 4 (plus slack), even
#define MTILES  625         // 10000 / 16

// workspace layout (float offsets)
#define OFF_Y     0          // 10000*48  = 480000
#define OFF_AGG1  480000     // 10000*16  = 160000
#define OFF_DEG   640000     // 10000
#define OFF_Z     650000     // 10000*24  = 240000  (z0:[0..6] z1:[8..14] zr:[16..22])
#define OFF_AGG2  890000     // 10000*8   = 80000
// total: 970000 floats = 3.88 MB

typedef float v2f __attribute__((ext_vector_type(2)));
typedef float v8f __attribute__((ext_vector_type(8)));

// ------------------------------------------------------------------
// Kernel 1: fused GEMM  Y[10000,48] = X[10000,1433] @ Wcat[1433,48]
// Wcat staged transposed + zero-padded in LDS (48 x 1440 f32 = 276.5 KB).
// One wave per 16-row tile; 8 waves / workgroup share the LDS weights.
// f32 WMMA 16x16x4, 3 N-tiles accumulated per wave.
// ------------------------------------------------------------------
__global__ __launch_bounds__(256) void k_gemm1(const float* __restrict__ x,
                                               const float* __restrict__ w1s,
                                               const float* __restrict__ w1r,
                                               float* __restrict__ Y) {
    extern __shared__ float lds[];   // [48][KPAD], K-major per column
    const int tid = threadIdx.x;

    // cooperative transpose+pad of the 1433x48 weight block into LDS
    for (int idx = tid; idx < NCAT * KPAD; idx += 256) {
        int n = idx / KPAD;
        int k = idx - n * KPAD;
        float v = 0.0f;
        if (k < IN_CH) {
            if (n < 16)      v = w1s[k * 16 + n];                 // W1_spline[0]
            else if (n < 32) v = w1s[IN_CH * 16 + k * 16 + (n - 16)]; // W1_spline[1]
            else             v = w1r[k * 16 + (n - 32)];          // W1_root
        }
        lds[idx] = v;
    }
    __syncthreads();

    const int wave = tid >> 5;
    const int lane = tid & 31;
    const int t    = blockIdx.x * 8 + wave;     // M-tile id (wave-uniform)
    if (t >= MTILES) return;                    // uniform per wave -> EXEC all-1s

    const int col  = lane & 15;
    const int half = lane >> 4;                 // K-half select per ISA layout
    const int row  = t * 16 + col;              // A-matrix row for this lane
    const float* xr = x + (size_t)row * IN_CH;

    const float* b0p = &lds[(0 * 16 + col) * KPAD];
    const float* b1p = &lds[(1 * 16 + col) * KPAD];
    const float* b2p = &lds[(2 * 16 + col) * KPAD];

    v8f acc0 = {}, acc1 = {}, acc2 = {};

    // main loop: 358 steps, k <= 1431 strictly in-bounds
    int kb = 0;
    for (; kb < 1432; kb += 4) {
        const int k0 = kb + half * 2;
        v2f a;
        a.x = xr[k0];
        a.y = xr[k0 + 1];
        v2f b0 = *(const v2f*)(b0p + k0);   // 8B-aligned ds_load_b64
        v2f b1 = *(const v2f*)(b1p + k0);
        v2f b2 = *(const v2f*)(b2p + k0);
        acc0 = __builtin_amdgcn_wmma_f32_16x16x4_f32(false, a, false, b0, (short)0, acc0, false, false);
        acc1 = __builtin_amdgcn_wmma_f32_16x16x4_f32(false, a, false, b1, (short)0, acc1, false, false);
        acc2 = __builtin_amdgcn_wmma_f32_16x16x4_f32(false, a, false, b2, (short)0, acc2, false, false);
    }
    // tail step kb == 1432: clamp A addresses (garbage lanes hit zero-padded B)
    {
        const int k0 = kb + half * 2;              // 1432 or 1434
        const int ka = (k0     < IN_CH) ? k0       : (IN_CH - 1);
        const int kc = (k0 + 1 < IN_CH) ? (k0 + 1) : (IN_CH - 1);
        v2f a;
        a.x = xr[ka];
        a.y = xr[kc];
        v2f b0 = *(const v2f*)(b0p + k0);
        v2f b1 = *(const v2f*)(b1p + k0);
        v2f b2 = *(const v2f*)(b2p + k0);
        acc0 = __builtin_amdgcn_wmma_f32_16x16x4_f32(false, a, false, b0, (short)0, acc0, false, false);
        acc1 = __builtin_amdgcn_wmma_f32_16x16x4_f32(false, a, false, b1, (short)0, acc1, false, false);
        acc2 = __builtin_amdgcn_wmma_f32_16x16x4_f32(false, a, false, b2, (short)0, acc2, false, false);
    }

    // C/D layout: VGPR v -> M = v + 8*half, N = lane&15
    const int rbase = t * 16 + half * 8;
    #pragma unroll
    for (int v = 0; v < 8; ++v) {
        float* yrow = Y + (size_t)(rbase + v) * NCAT;
        yrow[ 0 + col] = acc0[v];
        yrow[16 + col] = acc1[v];
        yrow[32 + col] = acc2[v];
    }
}

// ------------------------------------------------------------------
// Kernel 2: layer-1 edge scatter.  msg = (1-u)*Y0[src] + u*Y1[src]
// one thread per (edge, channel); f32 global atomics into agg1; deg count.
// ------------------------------------------------------------------
__global__ __launch_bounds__(256) void k_edge1(const int* __restrict__ ei,
                                               const float* __restrict__ pseudo,
                                               const float* __restrict__ Y,
                                               float* __restrict__ agg1,
                                               float* __restrict__ deg) {
    const int tid = blockIdx.x * 256 + threadIdx.x;
    if (tid >= NE * 16) return;
    const int e = tid >> 4;
    const int c = tid & 15;
    const int s = ei[e];
    const int d = ei[NE + e];
    const float u = pseudo[e];
    const float* ys = Y + (size_t)s * NCAT;
    const float m = (1.0f - u) * ys[c] + u * ys[16 + c];
    atomicAdd(&agg1[(size_t)d * 16 + c], m);
    if (c == 0) atomicAdd(&deg[d], 1.0f);
}

// ------------------------------------------------------------------
// Kernel 3: layer-1 finalize (normalize + root + bias + ELU) and the
// tiny layer-2 node transforms  z0/z1/zr = h1 @ {W2s0, W2s1, W2root}.
// ------------------------------------------------------------------
__global__ __launch_bounds__(256) void k_node1(const float* __restrict__ Y,
                                               const float* __restrict__ agg1,
                                               const float* __restrict__ deg,
                                               const float* __restrict__ b1,
                                               const float* __restrict__ w2s,
                                               const float* __restrict__ w2r,
                                               float* __restrict__ Z) {
    const int n = blockIdx.x * 256 + threadIdx.x;
    if (n >= NN) return;
    const float inv = 1.0f / fmaxf(deg[n], 1.0f);
    const float* ar = agg1 + (size_t)n * 16;
    const float* yr = Y + (size_t)n * NCAT + 32;

    float h[16];
    #pragma unroll
    for (int c = 0; c < 16; ++c) {
        const float v = ar[c] * inv + yr[c] + b1[c];
        h[c] = (v > 0.0f) ? v : expm1f(v);   // ELU, alpha=1
    }

    float z0[OUTC] = {}, z1[OUTC] = {}, zr[OUTC] = {};
    #pragma unroll
    for (int c = 0; c < 16; ++c) {
        const float hv = h[c];
        #pragma unroll
        for (int o = 0; o < OUTC; ++o) {
            z0[o] = fmaf(hv, w2s[c * OUTC + o], z0[o]);
            z1[o] = fmaf(hv, w2s[HID * OUTC + c * OUTC + o], z1[o]);
            zr[o] = fmaf(hv, w2r[c * OUTC + o], zr[o]);
        }
    }
    float* zn = Z + (size_t)n * 24;
    #pragma unroll
    for (int o = 0; o < OUTC; ++o) {
        zn[o] = z0[o]; zn[8 + o] = z1[o]; zn[16 + o] = zr[o];
    }
}

// ------------------------------------------------------------------
// Kernel 4: layer-2 edge scatter
// ------------------------------------------------------------------
__global__ __launch_bounds__(256) void k_edge2(const int* __restrict__ ei,
                                               const float* __restrict__ pseudo,
                                               const float* __restrict__ Z,
                                               float* __restrict__ agg2) {
    const int tid = blockIdx.x * 256 + threadIdx.x;
    if (tid >= NE * 8) return;
    const int c = tid & 7;
    if (c >= OUTC) return;
    const int e = tid >> 3;
    const int s = ei[e];
    const int d = ei[NE + e];
    const float u = pseudo[e];
    const float* zs = Z + (size_t)s * 24;
    const float m = (1.0f - u) * zs[c] + u * zs[8 + c];
    atomicAdd(&agg2[(size_t)d * 8 + c], m);
}

// ------------------------------------------------------------------
// Kernel 5: layer-2 finalize + log_softmax
// ------------------------------------------------------------------
__global__ __launch_bounds__(256) void k_out(const float* __restrict__ Z,
                                             const float* __restrict__ agg2,
                                             const float* __restrict__ deg,
                                             const float* __restrict__ b2,
                                             float* __restrict__ out) {
    const int n = blockIdx.x * 256 + threadIdx.x;
    if (n >= NN) return;
    const float inv = 1.0f / fmaxf(deg[n], 1.0f);
    const float* a  = agg2 + (size_t)n * 8;
    const float* zr = Z + (size_t)n * 24 + 16;
    float v[OUTC];
    float mx = -1e30f;
    #pragma unroll
    for (int c = 0; c < OUTC; ++c) {
        v[c] = a[c] * inv + zr[c] + b2[c];
        mx = fmaxf(mx, v[c]);
    }
    float ssum = 0.0f;
    #pragma unroll
    for (int c = 0; c < OUTC; ++c) ssum += expf(v[c] - mx);
    const float lse = mx + logf(ssum);
    float* o = out + (size_t)n * OUTC;
    #pragma unroll
    for (int c = 0; c < OUTC; ++c) o[c] = v[c] - lse;
}

// ------------------------------------------------------------------
extern "C" void kernel_launch(void* const* d_in, const int* in_sizes, int n_in,
                              void* d_out, int out_size, void* d_ws, size_t ws_size,
                              hipStream_t stream) {
    const float* x      = (const float*)d_in[0];
    const int*   ei     = (const int*)  d_in[1];
    const float* pseudo = (const float*)d_in[2];
    const float* w1s    = (const float*)d_in[3];
    const float* w1r    = (const float*)d_in[4];
    const float* b1     = (const float*)d_in[5];
    const float* w2s    = (const float*)d_in[6];
    const float* w2r    = (const float*)d_in[7];
    const float* b2     = (const float*)d_in[8];

    float* ws   = (float*)d_ws;
    float* Y    = ws + OFF_Y;
    float* agg1 = ws + OFF_AGG1;
    float* deg  = ws + OFF_DEG;
    float* Z    = ws + OFF_Z;
    float* agg2 = ws + OFF_AGG2;
    float* out  = (float*)d_out;

    // zero the accumulators every call (agg1 + deg are contiguous)
    hipMemsetAsync(agg1, 0, (160000 + 10000) * sizeof(float), stream);
    hipMemsetAsync(agg2, 0, 80000 * sizeof(float), stream);

    // 276.5 KB dynamic LDS (gfx1250 WGP supports up to 320 KB)
    const int ldsBytes = NCAT * KPAD * (int)sizeof(float);
    (void)hipFuncSetAttribute(reinterpret_cast<const void*>(k_gemm1),
                              hipFuncAttributeMaxDynamicSharedMemorySize, ldsBytes);

    k_gemm1<<<(MTILES + 7) / 8, 256, ldsBytes, stream>>>(x, w1s, w1r, Y);
    k_edge1<<<(NE * 16) / 256, 256, 0, stream>>>(ei, pseudo, Y, agg1, deg);
    k_node1<<<(NN + 255) / 256, 256, 0, stream>>>(Y, agg1, deg, b1, w2s, w2r, Z);
    k_edge2<<<(NE * 8) / 256, 256, 0, stream>>>(ei, pseudo, Z, agg2);
    k_out  <<<(NN + 255) / 256, 256, 0, stream>>>(Z, agg2, deg, b2, out);
}